// ConLoss_88390426951927
// MI455X (gfx1250) — compile-verified
//
#include <hip/hip_runtime.h>
#include <hip/hip_bf16.h>

#define BDIM 512
#define TJ   16
#define LPAD 4
#define LW   (BDIM + LPAD)   // 516: padded LDS row (2064B, 16B-aligned) -> conflict-free column gathers

typedef __attribute__((ext_vector_type(2))) float v2f;
typedef __attribute__((ext_vector_type(8))) float v8f;

// ---------------------------------------------------------------------------
// ws float layout:
//   [0] = cntPos, [1] = cntNeg
//   [16 .. 16+511]    = cls[b] in {+1,-1}
//   [528 .. 528+511]  = partial sums, fused axis1+axis2 pass (per i, 512 entries)
//   [1040.. 1040+127] = partial sums, axis0 pass (per 4-row block, 128 entries)
// ---------------------------------------------------------------------------

__device__ __forceinline__ float block_reduce_sum_512(float v, float* red) {
    const int lane = threadIdx.x & 31;
    const int w    = threadIdx.x >> 5;
    #pragma unroll
    for (int off = 16; off >= 1; off >>= 1) v += __shfl_xor(v, off, 32);
    if (lane == 0) red[w] = v;
    __syncthreads();
    float r = 0.f;
    if (threadIdx.x < 16) {
        r = red[threadIdx.x];
        #pragma unroll
        for (int off = 8; off >= 1; off >>= 1) r += __shfl_xor(r, off, 32);
    }
    return r;  // valid in thread 0
}

// Classify targets (round(t) >= 0 -> +1 else -1) and count positives.
__global__ __launch_bounds__(BDIM) void k_setup(const float* __restrict__ target,
                                                float* __restrict__ ws) {
    __shared__ float red[16];
    const int t = threadIdx.x;
    const float c = (rintf(target[t]) >= 0.f) ? 1.f : -1.f;
    ws[16 + t] = c;
    float p = (c > 0.f) ? 1.f : 0.f;
    float tot = block_reduce_sum_512(p, red);
    if (t == 0) { ws[0] = tot; ws[1] = (float)BDIM - tot; }
}

// Fused pass: one workgroup per i. Streams slice x[i,:,:] (1 MB) exactly once.
//  - axis 1 (softmax over j): thread t owns column k=t; running sum(exp) + masked raw sum.
//  - axis 2 (softmax over k): per 16-row LDS tile, WMMA f32 16x16x4 row-sum reduction
//    (A = data fragment, B = broadcast mask / ones columns), 16 waves own disjoint k-chunks.
__global__ __launch_bounds__(BDIM) void k_ab(const float* __restrict__ x,
                                             const float* __restrict__ ws,
                                             float* __restrict__ partialAB) {
    __shared__ __attribute__((aligned(16))) float raw[TJ][LW];
    __shared__ float mcls[BDIM];
    __shared__ float pm[16][16];   // per-wave partial masked row sums
    __shared__ float pe[16][16];   // per-wave partial exp row sums
    __shared__ float red[16];

    const int i    = blockIdx.x;
    const int t    = threadIdx.x;
    const int lane = t & 31;
    const int w    = t >> 5;

    const float ci      = ws[16 + i];
    const float cnt     = (ci > 0.f) ? ws[0] : ws[1];
    const float inv_cnt = 1.f / cnt;

    mcls[t] = (ws[16 + t] == ci) ? 1.f : 0.f;
    __syncthreads();

    float sB   = 0.f;  // sum_j exp(x[i,j,t])
    float smB  = 0.f;  // sum_{j: c[j]==ci} x[i,j,t]
    float accA = 0.f;  // axis-2 contributions (threads 0..15)

    const float* xi = x + (size_t)i * BDIM * BDIM;
    const int M  = lane & 15;
    const int kh = lane >> 4;

    for (int tile = 0; tile < BDIM / TJ; ++tile) {
        const int j0 = tile * TJ;

        // Stage 16 rows (32 KB) into LDS with wide b128 loads/stores:
        // tile = 2048 float4s, 4 per thread, fully coalesced.
        const float4* xt = (const float4*)(xi + (size_t)j0 * BDIM);
        #pragma unroll
        for (int q = 0; q < 4; ++q) {
            const int f = t + BDIM * q;          // 0..2047
            const float4 v = xt[f];
            const int r = f >> 7;                // row within tile
            const int c = (f & 127) << 2;        // column (multiple of 4)
            *(float4*)&raw[r][c] = v;
        }
        __syncthreads();

        // axis-1 running sums (column t).
        #pragma unroll 4
        for (int r = 0; r < TJ; ++r) {
            const float xv = raw[r][t];
            sB  += __expf(xv);
            smB += mcls[j0 + r] * xv;
        }

        // axis-2 row sums via WMMA: wave w handles k-chunks c = w + 16*cc.
        v8f Cm = {};  // accumulates sum_{k: c[k]==ci} x  (per row, all N columns equal)
        v8f Ce = {};  // accumulates sum_k exp(x)
        #pragma unroll
        for (int cc = 0; cc < 8; ++cc) {
            const int kk = 4 * (w + 16 * cc) + 2 * kh;
            v2f Ar = { raw[M][kk], raw[M][kk + 1] };
            v2f Ae = { __expf(Ar.x), __expf(Ar.y) };
            v2f Bm = { mcls[kk], mcls[kk + 1] };
            v2f Bo = { 1.f, 1.f };
            Cm = __builtin_amdgcn_wmma_f32_16x16x4_f32(false, Ar, false, Bm,
                                                       (short)0, Cm, false, false);
            Ce = __builtin_amdgcn_wmma_f32_16x16x4_f32(false, Ae, false, Bo,
                                                       (short)0, Ce, false, false);
        }
        // Extract column N=0 of each wave's C: lane 0 holds rows 0-7, lane 16 rows 8-15.
        if (lane == 0 || lane == 16) {
            const int rb = (lane == 0) ? 0 : 8;
            #pragma unroll
            for (int q = 0; q < 8; ++q) { pm[w][rb + q] = Cm[q]; pe[w][rb + q] = Ce[q]; }
        }
        __syncthreads();

        // Combine 16 wave partials; accumulate axis-2 loss for matched rows.
        if (t < TJ) {
            float ssame = 0.f, sexp = 0.f;
            #pragma unroll
            for (int ww = 0; ww < 16; ++ww) { ssame += pm[ww][t]; sexp += pe[ww][t]; }
            if (mcls[j0 + t] != 0.f)
                accA += __logf(sexp) - ssame * inv_cnt;
        }
        __syncthreads();
    }

    float contrib = accA;
    if (mcls[t] != 0.f)                       // axis-1 pair (i, k=t) matched
        contrib += __logf(sB) - smB * inv_cnt;

    const float tot = block_reduce_sum_512(contrib, red);
    if (t == 0) partialAB[i] = tot;
}

// axis 0 (softmax over i): one workgroup per 4 consecutive j rows.
// Per i iteration the block streams one contiguous 8 KB region x[i, j0:j0+4, :]
// with one float4 (global_load_b128) per thread; each thread owns 4 columns.
__global__ __launch_bounds__(BDIM) void k_c(const float* __restrict__ x,
                                            const float* __restrict__ ws,
                                            float* __restrict__ partialC) {
    __shared__ float clsh[BDIM];
    __shared__ float red[16];
    const int t  = threadIdx.x;
    const int j0 = blockIdx.x * 4;
    clsh[t] = ws[16 + t];
    __syncthreads();

    const int   jj  = t >> 7;            // which of the 4 rows this thread serves
    const float cj  = clsh[j0 + jj];
    const float cnt = (cj > 0.f) ? ws[0] : ws[1];

    float4 se = {0.f, 0.f, 0.f, 0.f};
    float4 sm = {0.f, 0.f, 0.f, 0.f};
    for (int i = 0; i < BDIM; ++i) {
        const float4* p = (const float4*)(x + ((size_t)i * BDIM + j0) * BDIM);
        const float4 v = p[t];
        const float wi = (clsh[i] == cj) ? 1.f : 0.f;
        se.x += __expf(v.x); se.y += __expf(v.y);
        se.z += __expf(v.z); se.w += __expf(v.w);
        sm.x += wi * v.x; sm.y += wi * v.y;
        sm.z += wi * v.z; sm.w += wi * v.w;
    }

    const int   k0  = (t & 127) << 2;
    const float inv = 1.f / cnt;
    float contrib = 0.f;
    contrib += (clsh[k0 + 0] == cj) ? (__logf(se.x) - sm.x * inv) : 0.f;
    contrib += (clsh[k0 + 1] == cj) ? (__logf(se.y) - sm.y * inv) : 0.f;
    contrib += (clsh[k0 + 2] == cj) ? (__logf(se.z) - sm.z * inv) : 0.f;
    contrib += (clsh[k0 + 3] == cj) ? (__logf(se.w) - sm.w * inv) : 0.f;

    const float tot = block_reduce_sum_512(contrib, red);
    if (t == 0) partialC[blockIdx.x] = tot;
}

// Deterministic final reduction of 512 + 128 partials -> mean.
__global__ __launch_bounds__(BDIM) void k_final(const float* __restrict__ pAB,
                                                const float* __restrict__ pC,
                                                float* __restrict__ out) {
    __shared__ float red[16];
    const int t = threadIdx.x;
    float v = pAB[t] + ((t < 128) ? pC[t] : 0.f);
    const float tot = block_reduce_sum_512(v, red);
    if (t == 0) out[0] = tot * (1.f / ((float)BDIM * (float)BDIM));
}

extern "C" void kernel_launch(void* const* d_in, const int* in_sizes, int n_in,
                              void* d_out, int out_size, void* d_ws, size_t ws_size,
                              hipStream_t stream) {
    const float* cube   = (const float*)d_in[0];
    const float* target = (const float*)d_in[1];
    float* W   = (float*)d_ws;
    float* pAB = W + 528;
    float* pC  = W + 1040;
    float* out = (float*)d_out;

    k_setup<<<1, BDIM, 0, stream>>>(target, W);
    k_ab   <<<BDIM, BDIM, 0, stream>>>(cube, W, pAB);
    k_c    <<<128, BDIM, 0, stream>>>(cube, W, pC);
    k_final<<<1, BDIM, 0, stream>>>(pAB, pC, out);
}